// NLLLoss_customized_17583596110357
// MI455X (gfx1250) — compile-verified
//
#include <hip/hip_runtime.h>
#include <math.h>

// MI455X (gfx1250): wave32, WMMA. This kernel is HBM-bound:
// 537MB read-once / 23.3 TB/s ~= 23us floor. Design = coalesced b128
// non-temporal streaming, one wave per 128-float row, log2-space math
// (1 trans + 3 VALU per element), deterministic 2-pass reduction.
// Final 32-lane sum uses V_WMMA_F32_16X16X4_F32 (layout-permutation-proof).

typedef float v2f __attribute__((ext_vector_type(2)));
typedef float v8f __attribute__((ext_vector_type(8)));
typedef float f4  __attribute__((ext_vector_type(4)));

#define NBLK 2048
#define TPB  256
#define WPB  8   // waves per block (wave32)

__device__ __forceinline__ void combineMax(float& m, int& mi, float om, int oi) {
  // max with tie -> smaller index == jnp.argmax "first occurrence" semantics
  if (om > m) { m = om; mi = oi; }
  else if (om == m && oi < mi) { mi = oi; }
}

__global__ __launch_bounds__(TPB) void nll_pass1(
    const float* __restrict__ x, const float* __restrict__ y,
    float* __restrict__ bsum, float* __restrict__ bmax, int* __restrict__ bidx,
    int rows)
{
  const int tid  = threadIdx.x;
  const int lane = tid & 31;
  const int wib  = tid >> 5;
  const int gw   = blockIdx.x * WPB + wib;     // global wave id
  const int tw   = NBLK * WPB;                 // total waves

  float ws = 0.0f;           // wave running sum of row losses (log2 units)
  float wm = -INFINITY;      // wave running max
  int   wi = 2147483647;     // its row index

  for (int row = gw; row < rows; row += tw) {
    size_t off = (size_t)row * 128u + (size_t)(lane << 2);
    // read-once stream: TH=NT, keep out of L2 (537MB > 192MB L2)
    f4 xv = __builtin_nontemporal_load((const f4*)(x + off));
    f4 yv = __builtin_nontemporal_load((const f4*)(y + off));
    // t = y*log2(x) <= 0 always (y>=0, x<=1); fminf(t,0): NaN -> 0 (minNum),
    // finite t passes through exactly  ==  nansum semantics, 1 VALU op.
    float t0 = fminf(yv.x * __log2f(xv.x), 0.0f);
    float t1 = fminf(yv.y * __log2f(xv.y), 0.0f);
    float t2 = fminf(yv.z * __log2f(xv.z), 0.0f);
    float t3 = fminf(yv.w * __log2f(xv.w), 0.0f);
    float p = (t0 + t1) + (t2 + t3);
    // 32-lane butterfly: all lanes end with the full row sum
    #pragma unroll
    for (int o = 16; o > 0; o >>= 1) p += __shfl_xor(p, o, 32);
    float rs = -p;                    // row loss (log2 units, >= 0)
    ws += rs;                         // identical on all lanes
    if (rs > wm) { wm = rs; wi = row; }   // strict '>' keeps first max
  }

  __shared__ float sS[WPB];
  __shared__ float sM[WPB];
  __shared__ int   sI[WPB];
  if (lane == 0) { sS[wib] = ws; sM[wib] = wm; sI[wib] = wi; }
  __syncthreads();
  if (tid == 0) {
    float bs = 0.0f, bm = -INFINITY; int bi = 2147483647;
    #pragma unroll
    for (int i = 0; i < WPB; ++i) { bs += sS[i]; combineMax(bm, bi, sM[i], sI[i]); }
    bsum[blockIdx.x] = bs; bmax[blockIdx.x] = bm; bidx[blockIdx.x] = bi;
  }
}

__global__ __launch_bounds__(256) void nll_pass2(
    const float* __restrict__ bsum, const float* __restrict__ bmax,
    const int* __restrict__ bidx, float* __restrict__ out, int rows)
{
  const int tid = threadIdx.x;
  __shared__ float sS[256];
  __shared__ float sM[256];
  __shared__ int   sI[256];

  float s = 0.0f, m = -INFINITY; int mi = 2147483647;
  for (int j = tid; j < NBLK; j += 256) {          // fixed order: deterministic
    s += bsum[j];
    combineMax(m, mi, bmax[j], bidx[j]);
  }
  sS[tid] = s; sM[tid] = m; sI[tid] = mi;
  __syncthreads();

  if (tid < 32) {                                   // wave 0, EXEC all-ones
    float v = 0.0f; float m2 = -INFINITY; int mi2 = 2147483647;
    #pragma unroll
    for (int k = 0; k < 8; ++k) {
      int j = tid * 8 + k;
      v += sS[j];
      combineMax(m2, mi2, sM[j], sI[j]);
    }
    // max/argmax butterfly (assoc+comm combine -> all lanes get global result)
    #pragma unroll
    for (int o = 16; o > 0; o >>= 1) {
      float om = __shfl_xor(m2, o, 32);
      int   oi = __shfl_xor(mi2, o, 32);
      combineMax(m2, mi2, om, oi);
    }
    // Sum the 32 per-lane partials with V_WMMA_F32_16X16X4_F32:
    // A = ones(16x4) => D[m][n] = sum_k B[k][n]. The 16 column sums
    // partition all 64 B slots ({b.x,b.y} x 32 lanes), zeros in b.y add 0.
    // C/D layout: d[0] holds each column sum exactly twice across the wave
    // (M=0 rows in lanes 0-15, identical M=8 rows in lanes 16-31),
    // so 0.5 * lane-sum(d[0]) == total, for ANY B slot permutation.
    v2f a; a.x = 1.0f; a.y = 1.0f;
    v2f b; b.x = v;    b.y = 0.0f;
    v8f c = {};
    v8f d = __builtin_amdgcn_wmma_f32_16x16x4_f32(
        /*neg_a=*/false, a, /*neg_b=*/false, b,
        /*c_mod=*/(short)0, c, /*reuse_a=*/false, /*reuse_b=*/false);
    float t = d[0];
    #pragma unroll
    for (int o = 16; o > 0; o >>= 1) t += __shfl_xor(t, o, 32);
    float total = t * 0.5f;   // exact halving of exact doubling

    if (tid == 0) {
      const double LN2 = 0.6931471805599453094;   // convert log2 -> ln once
      out[0] = (float)((double)total * LN2 / (double)rows);  // mean loss
      out[1] = (float)((double)m2 * LN2);                    // max loss
      out[2] = (float)mi2;                                   // argmax row
    }
  }
}

extern "C" void kernel_launch(void* const* d_in, const int* in_sizes, int n_in,
                              void* d_out, int out_size, void* d_ws, size_t ws_size,
                              hipStream_t stream) {
  const float* x = (const float*)d_in[0];
  const float* y = (const float*)d_in[1];
  int rows = in_sizes[0] / 128;

  float* ws0  = (float*)d_ws;
  float* bsum = ws0;                 // [NBLK]
  float* bmax = ws0 + NBLK;          // [NBLK]
  int*   bidx = (int*)(ws0 + 2 * NBLK); // [NBLK]  (24 KB total scratch)

  nll_pass1<<<NBLK, TPB, 0, stream>>>(x, y, bsum, bmax, bidx, rows);
  nll_pass2<<<1, 256, 0, stream>>>(bsum, bmax, bidx, (float*)d_out, rows);
}